// OfficialMamba3SISO_86062554677279
// MI455X (gfx1250) — compile-verified
//
#include <hip/hip_runtime.h>
#include <hip/hip_bf16.h>
#include <math.h>

#define D_MODEL    1024
#define D_STATE    128
#define HEADDIM    64
#define NHEADS     32
#define D_INNER    2048
#define D_IN_PROJ  4480
#define TOKENS     2048   // b * l
#define SEQLEN     1024
#define Q          64     // chunk length
#define NCH        16     // SEQLEN / Q
#define BH         64     // b * NHEADS
#define A_FLOOR    1e-4f
#define EPS_RMS    1e-5f

typedef __attribute__((ext_vector_type(2))) float v2f;
typedef __attribute__((ext_vector_type(8))) float v8f;

__device__ inline float softplus_f(float x) {
  return x > 20.f ? x : log1pf(expf(x));
}

__device__ inline v8f wmma4(v2f a, v2f b, v8f c) {
  return __builtin_amdgcn_wmma_f32_16x16x4_f32(false, a, false, b, (short)0, c,
                                               false, false);
}

// ---------------------------------------------------------------------------
// NT GEMM tile: C[16,16] += A[16xK] * B[16xK]^T, both row-major K-contiguous.
// f32 WMMA layout: lane&15 = row, (lane>>4)*2+v = K slot; C VGPR r ->
// row = tm + r + (lane>>4)*8, col = tn + (lane&15).
// ---------------------------------------------------------------------------
template<int K>
__device__ inline void wmma_tile_nt(const float* __restrict__ A,
                                    const float* __restrict__ B,
                                    float* __restrict__ C, int ldc,
                                    int tm, int tn) {
  const int lane = threadIdx.x & 31;
  const int rlo  = lane & 15;
  const int hi   = lane >> 4;
  const float* arow = A + (size_t)(tm + rlo) * K;
  const float* brow = B + (size_t)(tn + rlo) * K;
  v8f acc = {};
  #pragma unroll 8
  for (int k = 0; k < K; k += 4) {
    const int kk = k + hi * 2;
    v2f a = *(const v2f*)(arow + kk);
    v2f b = *(const v2f*)(brow + kk);
    acc = wmma4(a, b, acc);
  }
  const int row0 = tm + hi * 8;
  const int col  = tn + rlo;
  #pragma unroll
  for (int r = 0; r < 8; ++r)
    C[(size_t)(row0 + r) * ldc + col] = acc[r];
}

__global__ __launch_bounds__(128)
void proj_gemm(const float* __restrict__ U, const float* __restrict__ Win,
               float* __restrict__ P) {
  const int wave = threadIdx.x >> 5;
  const int tn = (blockIdx.x * 4 + wave) * 16;
  const int tm = blockIdx.y * 16;
  wmma_tile_nt<D_MODEL>(U, Win, P, D_IN_PROJ, tm, tn);
}

__global__ __launch_bounds__(128)
void out_gemm(const float* __restrict__ Y, const float* __restrict__ Wout,
              float* __restrict__ O) {
  const int wave = threadIdx.x >> 5;
  const int tn = (blockIdx.x * 4 + wave) * 16;
  const int tm = blockIdx.y * 16;
  wmma_tile_nt<D_INNER>(Y, Wout, O, D_MODEL, tm, tn);
}

// ---------------------------------------------------------------------------
// Per-(b,h) scalar coefficients + sequential cumsums (theta over full seq,
// s over each chunk). One wave per (b,h); lane i owns rope angle i.
// proj row layout: z@0 x@2048 B@4096 C@4224 dd_dt@4352 dd_A@4384 trap@4416
//                  angles@4448
// ---------------------------------------------------------------------------
__global__ __launch_bounds__(32)
void coef_scan(const float* __restrict__ P, const float* __restrict__ dt_bias,
               float* __restrict__ theta, float* __restrict__ sloc,
               float* __restrict__ alph, float* __restrict__ betav,
               float* __restrict__ dl) {
  const int bh = blockIdx.x;
  const int batch = bh >> 5, h = bh & 31;
  const int lane = threadIdx.x;
  const float dtb = dt_bias[h];
  float th_run = 0.f, s_run = 0.f;
  for (int t = 0; t < SEQLEN; ++t) {
    const size_t base = (size_t)(batch * SEQLEN + t) * D_IN_PROJ;
    const float ddt = P[base + 4352 + h];
    const float dda = P[base + 4384 + h];
    const float trp = P[base + 4416 + h];
    const float dt  = softplus_f(ddt + dtb);
    float a = -softplus_f(dda);
    a = fminf(a, -A_FLOOR);
    const float adt = a * dt;
    const float lam = 1.f / (1.f + expf(-trp));
    const float g   = expf(adt);
    s_run = ((t & (Q - 1)) == 0) ? adt : (s_run + adt);
    th_run += P[base + 4448 + lane] * dt;
    const size_t idx = (size_t)bh * SEQLEN + t;
    theta[idx * 32 + lane] = th_run;
    if (lane == 0) {
      sloc[idx]  = s_run;
      alph[idx]  = dt * lam;
      betav[idx] = dt * (1.f - lam) * g;
      dl[idx]    = dt * (1.f - lam);
    }
  }
}

// ---------------------------------------------------------------------------
// Stage rms-normed + biased + rope'd B or C for one chunk into LDS [64][128].
// 4 lanes per timestep: lane j0 owns n = j0+4i; the rope partner (n, n+32)
// stays within the same lane's stride-4 set.
// ---------------------------------------------------------------------------
__device__ inline void stage_rope(const float* __restrict__ P,
                                  const float* __restrict__ theta,
                                  const float* __restrict__ w,
                                  const float* __restrict__ bias,
                                  float* sOut, int batch, int h, int c,
                                  int srcOff) {
  const int tid = threadIdx.x;
  const int tl  = tid >> 2;      // local t: 0..63
  const int j0  = tid & 3;
  const int tt  = c * Q + tl;
  const size_t base = (size_t)(batch * SEQLEN + tt) * D_IN_PROJ + srcOff;
  float raw[32];
  float sq = 0.f;
  #pragma unroll
  for (int i = 0; i < 32; ++i) {
    const int n = j0 + 4 * i;
    const float v = P[base + n];
    raw[i] = v;
    sq += v * v;
  }
  sq += __shfl_xor(sq, 1);
  sq += __shfl_xor(sq, 2);
  const float rs = rsqrtf(sq * (1.f / 128.f) + EPS_RMS);
  const int bh = (batch << 5) | h;
  const float* th = theta + ((size_t)bh * SEQLEN + tt) * 32;
  #pragma unroll
  for (int i = 0; i < 32; ++i) {
    const int n = j0 + 4 * i;
    raw[i] = raw[i] * rs * w[n] + bias[h * 128 + n];
  }
  #pragma unroll
  for (int i = 0; i < 32; ++i) {
    const int n = j0 + 4 * i;
    float o;
    if (n < 32) {
      const float ct = cosf(th[n]), st = sinf(th[n]);
      o = raw[i] * ct - raw[i + 8] * st;
    } else if (n < 64) {
      const float ct = cosf(th[n - 32]), st = sinf(th[n - 32]);
      o = raw[i - 8] * st + raw[i] * ct;
    } else {
      o = raw[i];
    }
    sOut[tl * 128 + n] = o;
  }
}

// ---------------------------------------------------------------------------
// Per head-chunk state contribution: G'[p,n] = sum_j w'_j x_j[p] Bh_j[n]
// w'_j = W(Q-1, j) (+ next-chunk trapezoid correction on j = Q-1).
// ---------------------------------------------------------------------------
__global__ __launch_bounds__(256)
void chunk_state(const float* __restrict__ P, const float* __restrict__ theta,
                 const float* __restrict__ sloc, const float* __restrict__ alph,
                 const float* __restrict__ betav, const float* __restrict__ dl,
                 const float* __restrict__ Bw, const float* __restrict__ Bb,
                 float* __restrict__ Gp) {
  const int c = blockIdx.x, bh = blockIdx.y;
  const int batch = bh >> 5, h = bh & 31;
  const int tid = threadIdx.x;
  __shared__ float sBh[Q * 128];
  __shared__ float sXw[Q * Q];
  __shared__ float sw[Q];
  stage_rope(P, theta, Bw, Bb, sBh, batch, h, c, 4096);
  if (tid < Q) {
    const int j = tid;
    const size_t cb = (size_t)bh * SEQLEN + c * Q;
    const float sl = sloc[cb + Q - 1];
    float wv;
    if (j == Q - 1) {
      wv = alph[cb + Q - 1] + ((c < NCH - 1) ? dl[cb + Q] : 0.f);
    } else {
      wv = expf(sl - sloc[cb + j]) * alph[cb + j]
         + expf(sl - sloc[cb + j + 1]) * betav[cb + j + 1];
    }
    sw[j] = wv;
  }
  __syncthreads();
  for (int e = tid; e < Q * Q; e += 256) {
    const int p_ = e >> 6, j = e & 63;
    const size_t rb = (size_t)(batch * SEQLEN + c * Q + j) * D_IN_PROJ;
    sXw[p_ * Q + j] = P[rb + 2048 + h * 64 + p_] * sw[j];
  }
  __syncthreads();
  const int wave = tid >> 5, lane = tid & 31;
  const int rlo = lane & 15, hi = lane >> 4;
  float* Gout = Gp + ((size_t)bh * NCH + c) * (HEADDIM * D_STATE);
  // 4 p-tiles x 8 n-tiles = 32 tiles; A row-K (sXw), B column-N (sBh).
  for (int tile = wave; tile < 32; tile += 8) {
    const int tm = (tile >> 3) * 16;   // p
    const int tn = (tile & 7) * 16;    // n
    v8f acc = {};
    #pragma unroll 4
    for (int k = 0; k < Q; k += 4) {
      const int kk = k + hi * 2;
      v2f a = *(const v2f*)(&sXw[(tm + rlo) * Q + kk]);
      v2f b;
      b[0] = sBh[kk * 128 + tn + rlo];
      b[1] = sBh[(kk + 1) * 128 + tn + rlo];
      acc = wmma4(a, b, acc);
    }
    const int row0 = tm + hi * 8, col = tn + rlo;
    #pragma unroll
    for (int r = 0; r < 8; ++r) Gout[(row0 + r) * 128 + col] = acc[r];
  }
}

// ---------------------------------------------------------------------------
// Sequential combine over 16 chunks: Hhat_{c+1} = exp(T_c) Hhat_c + G'_c.
// Overwrites each G' slot with Hhat_c (the state entering chunk c).
// ---------------------------------------------------------------------------
__global__ __launch_bounds__(256)
void state_combine(const float* __restrict__ sloc, float* __restrict__ Gp) {
  const int bh = blockIdx.x, tid = threadIdx.x;
  float carry[32];
  #pragma unroll
  for (int i = 0; i < 32; ++i) carry[i] = 0.f;
  for (int c = 0; c < NCH; ++c) {
    float* slot = Gp + ((size_t)bh * NCH + c) * (HEADDIM * D_STATE) + tid;
    const float gT = expf(sloc[(size_t)bh * SEQLEN + c * Q + Q - 1]);
    #pragma unroll
    for (int i = 0; i < 32; ++i) {
      const float gv = slot[i * 256];
      slot[i * 256] = carry[i];          // publish Hhat_c
      carry[i] = gT * carry[i] + gv;
    }
  }
}

// ---------------------------------------------------------------------------
// Per head-chunk output:
//   S[t,j]   = (Ch_t . Bh_j) * W(t,j)            (masked, WMMA + elementwise)
//   Y[t,p]   = (S @ X)[t,p] + exp(s_t)*(Ch_t . Hhat[p,:]) + D*x ; silu(z) gate
// ---------------------------------------------------------------------------
__global__ __launch_bounds__(256)
void chunk_y(const float* __restrict__ P, const float* __restrict__ theta,
             const float* __restrict__ slocA, const float* __restrict__ alph,
             const float* __restrict__ betav,
             const float* __restrict__ Bw, const float* __restrict__ Bb,
             const float* __restrict__ Cw, const float* __restrict__ Cb,
             const float* __restrict__ Dp, const float* __restrict__ Gp,
             float* __restrict__ Y) {
  const int c = blockIdx.x, bh = blockIdx.y;
  const int batch = bh >> 5, h = bh & 31;
  const int tid = threadIdx.x;
  __shared__ float sBh[Q * 128], sCh[Q * 128];
  __shared__ float sS[Q * Q], sXt[Q * Q];
  __shared__ float ss[Q], sa[Q], sb[Q];
  stage_rope(P, theta, Bw, Bb, sBh, batch, h, c, 4096);
  stage_rope(P, theta, Cw, Cb, sCh, batch, h, c, 4224);
  if (tid < Q) {
    const size_t cb = (size_t)bh * SEQLEN + c * Q + tid;
    ss[tid] = slocA[cb];
    sa[tid] = alph[cb];
    sb[tid] = betav[cb];
  }
  for (int e = tid; e < Q * Q; e += 256) {
    const int p_ = e >> 6, j = e & 63;
    const size_t rb = (size_t)(batch * SEQLEN + c * Q + j) * D_IN_PROJ;
    sXt[p_ * Q + j] = P[rb + 2048 + h * 64 + p_];   // x_j[p] at [p][j]
  }
  __syncthreads();
  const int wave = tid >> 5, lane = tid & 31;
  const int rlo = lane & 15, hi = lane >> 4;
  // ---- phase S: S = Ch * Bh^T, masked ----
  for (int tile = wave; tile < 16; tile += 8) {
    const int tm = (tile >> 2) * 16, tn = (tile & 3) * 16;
    v8f acc = {};
    #pragma unroll 4
    for (int k = 0; k < 128; k += 4) {
      const int kk = k + hi * 2;
      v2f a = *(const v2f*)(&sCh[(tm + rlo) * 128 + kk]);
      v2f b = *(const v2f*)(&sBh[(tn + rlo) * 128 + kk]);
      acc = wmma4(a, b, acc);
    }
    const int row0 = tm + hi * 8, col = tn + rlo;
    #pragma unroll
    for (int r = 0; r < 8; ++r) {
      const int t = row0 + r, j = col;
      float m;
      if (j > t)       m = 0.f;
      else if (j == t) m = sa[j];
      else m = expf(ss[t] - ss[j]) * sa[j]
             + expf(ss[t] - ss[j + 1]) * sb[j + 1];
      sS[t * Q + j] = acc[r] * m;
    }
  }
  __syncthreads();
  // ---- phase Y: intra (S @ X) + inter (Ch . Hhat^T), gate, store ----
  const float* Hh = Gp + ((size_t)bh * NCH + c) * (HEADDIM * D_STATE);
  const float Dph = Dp[h];
  for (int tile = wave; tile < 16; tile += 8) {
    const int tm = (tile >> 2) * 16, tn = (tile & 3) * 16;  // t-tile, p-tile
    v8f acc = {};
    #pragma unroll 4
    for (int k = 0; k < Q; k += 4) {
      const int kk = k + hi * 2;
      v2f a = *(const v2f*)(&sS[(tm + rlo) * Q + kk]);
      v2f b = *(const v2f*)(&sXt[(tn + rlo) * Q + kk]);
      acc = wmma4(a, b, acc);
    }
    v8f acc2 = {};
    #pragma unroll 4
    for (int k = 0; k < 128; k += 4) {
      const int kk = k + hi * 2;
      v2f a = *(const v2f*)(&sCh[(tm + rlo) * 128 + kk]);
      v2f b = *(const v2f*)(Hh + (size_t)(tn + rlo) * 128 + kk);
      acc2 = wmma4(a, b, acc2);
    }
    const int row0 = tm + hi * 8, col = tn + rlo;
    #pragma unroll
    for (int r = 0; r < 8; ++r) {
      const int t = row0 + r, p_ = col;
      const size_t rb = (size_t)(batch * SEQLEN + c * Q + t) * D_IN_PROJ;
      const float z  = P[rb + h * 64 + p_];
      const float xv = sXt[p_ * Q + t];
      const float yv = acc[r] + expf(ss[t]) * acc2[r] + Dph * xv;
      const float sz = z / (1.f + expf(-z));
      Y[(size_t)(batch * SEQLEN + c * Q + t) * D_INNER + h * 64 + p_] = yv * sz;
    }
  }
}

extern "C" void kernel_launch(void* const* d_in, const int* in_sizes, int n_in,
                              void* d_out, int out_size, void* d_ws, size_t ws_size,
                              hipStream_t stream) {
  const float* u    = (const float*)d_in[0];
  const float* Win  = (const float*)d_in[1];
  const float* Wout = (const float*)d_in[2];
  const float* dtb  = (const float*)d_in[3];
  const float* Bb   = (const float*)d_in[4];
  const float* Cb   = (const float*)d_in[5];
  const float* Bw   = (const float*)d_in[6];
  const float* Cw   = (const float*)d_in[7];
  const float* Dp   = (const float*)d_in[8];
  float* out = (float*)d_out;

  // workspace map (floats)
  float* proj  = (float*)d_ws;                                   // 2048*4480
  float* theta = proj  + (size_t)TOKENS * D_IN_PROJ;             // 64*1024*32
  float* sloc  = theta + (size_t)BH * SEQLEN * 32;               // 64*1024
  float* alphv = sloc  + (size_t)BH * SEQLEN;
  float* betav = alphv + (size_t)BH * SEQLEN;
  float* dlv   = betav + (size_t)BH * SEQLEN;
  float* Gp    = dlv   + (size_t)BH * SEQLEN;                    // 1024*8192
  float* ybuf  = Gp    + (size_t)BH * NCH * HEADDIM * D_STATE;   // 2048*2048

  proj_gemm<<<dim3(D_IN_PROJ / 64, TOKENS / 16), 128, 0, stream>>>(u, Win, proj);
  coef_scan<<<BH, 32, 0, stream>>>(proj, dtb, theta, sloc, alphv, betav, dlv);
  chunk_state<<<dim3(NCH, BH), 256, 0, stream>>>(proj, theta, sloc, alphv,
                                                 betav, dlv, Bw, Bb, Gp);
  state_combine<<<BH, 256, 0, stream>>>(sloc, Gp);
  chunk_y<<<dim3(NCH, BH), 256, 0, stream>>>(proj, theta, sloc, alphv, betav,
                                             Bw, Bb, Cw, Cb, Dp, Gp, ybuf);
  out_gemm<<<dim3(D_MODEL / 64, TOKENS / 16), 128, 0, stream>>>(ybuf, Wout, out);
}